// GRUEncoder_5145370821090
// MI455X (gfx1250) — compile-verified
//
#include <hip/hip_runtime.h>
#include <hip/hip_bf16.h>
#include <math.h>

// ---------------------------------------------------------------------------
// GRU encoder:  V=32000, EH=512, EO=256, H=512, O=256, S=128, B=32
// fp32 end-to-end using V_WMMA_F32_16X16X4_F32 (exact fp32 semantics).
// Data movement into LDS uses gfx1250 async global->LDS (ASYNCcnt) path.
// ---------------------------------------------------------------------------

#define DV   32000
#define DEH  512
#define DEO  256
#define DH   512
#define DO_  256
#define DS_  128
#define DB   32
#define SB   (DS_ * DB)      // 4096
#define G3   (3 * DH)        // 1536
#define THRESH 1e-6f

typedef __attribute__((ext_vector_type(2))) float v2f;
typedef __attribute__((ext_vector_type(8))) float v8f;

__device__ __forceinline__ v8f wmma_f32(v2f a, v2f b, v8f c) {
  // D = A(16x4,f32) * B(4x16,f32) + C(16x16,f32)
  return __builtin_amdgcn_wmma_f32_16x16x4_f32(
      /*neg_a=*/false, a, /*neg_b=*/false, b,
      /*c_mod=*/(short)0, c, /*reuse_a=*/false, /*reuse_b=*/false);
}

// gfx1250 async global->LDS copy: 16B per lane, tracked with ASYNCcnt.
// vdst = per-lane LDS byte address, vaddr = per-lane 64-bit global address.
__device__ __forceinline__ void async_g2l_b128(unsigned lds_addr, const float* gsrc) {
  asm volatile("global_load_async_to_lds_b128 %0, %1, off"
               :: "v"(lds_addr), "v"(gsrc)
               : "memory");
}
__device__ __forceinline__ void wait_asynccnt0() {
  asm volatile("s_wait_asynccnt 0" ::: "memory");
}

// ---------------------------------------------------------------------------
// Kernel 1: embedding gather + b1 + threshold  -> e1 (4096 x 512)
// ---------------------------------------------------------------------------
__global__ void gather_embed_kernel(const int* __restrict__ tok,
                                    const float* __restrict__ W1,
                                    const float* __restrict__ b1,
                                    float* __restrict__ e1) {
  const int i = blockIdx.x;                 // token 0..4095 (flattened S,B)
  const int t = tok[i];
  const float* src = &W1[(size_t)t * DEH];
  float* dst = &e1[(size_t)i * DEH];
  for (int j = threadIdx.x; j < DEH; j += blockDim.x) {
    float v = src[j] + b1[j];
    dst[j] = (v > THRESH) ? v : 0.0f;
  }
}

// ---------------------------------------------------------------------------
// Generic fp32 WMMA GEMM: C[M,N] = act(A[M,K] @ B + bias)
//   TRANSB=0: B = W (K x N row-major)
//   TRANSB=1: B = W^T, W is (N x K) row-major
//   ACT=1: threshold(1e-6)
// One 16x16 tile per wave; 8 waves per block; M,N mult of 16, K mult of 8.
// ---------------------------------------------------------------------------
template <int TRANSB, int ACT>
__global__ void gemm_wmma_kernel(const float* __restrict__ A,
                                 const float* __restrict__ W,
                                 const float* __restrict__ bias,
                                 float* __restrict__ C,
                                 int M, int N, int K) {
  const int wave = blockIdx.x * (blockDim.x >> 5) + (threadIdx.x >> 5);
  const int ntiles = (M >> 4) * (N >> 4);
  if (wave >= ntiles) return;               // wave-uniform: EXEC stays all-ones
  const int ntn = N >> 4;
  const int mt = wave / ntn;
  const int nt = wave % ntn;

  const int lane = threadIdx.x & 31;
  const int l16  = lane & 15;
  const int kh   = (lane >> 4) << 1;        // K sub-offset {0,2}

  const int arow = mt * 16 + l16;           // A: lanes hold rows
  const int bcol = nt * 16 + l16;           // B/C: lanes hold cols
  const float* ap = &A[(size_t)arow * K + kh];

  v8f c0 = {}; v8f c1 = {};
  if (TRANSB) {
    const float* bp = &W[(size_t)bcol * K + kh];
    for (int k = 0; k < K; k += 8) {
      v2f a0 = { ap[k],     ap[k + 1] };
      v2f b0 = { bp[k],     bp[k + 1] };
      c0 = wmma_f32(a0, b0, c0);
      v2f a1 = { ap[k + 4], ap[k + 5] };
      v2f b1 = { bp[k + 4], bp[k + 5] };
      c1 = wmma_f32(a1, b1, c1);
    }
  } else {
    const float* bp = &W[(size_t)kh * N + bcol];
    for (int k = 0; k < K; k += 8) {
      v2f a0 = { ap[k],     ap[k + 1] };
      v2f b0 = { bp[(size_t)k * N],       bp[(size_t)(k + 1) * N] };
      c0 = wmma_f32(a0, b0, c0);
      v2f a1 = { ap[k + 4], ap[k + 5] };
      v2f b1 = { bp[(size_t)(k + 4) * N], bp[(size_t)(k + 5) * N] };
      c1 = wmma_f32(a1, b1, c1);
    }
  }

  const float bv = bias ? bias[bcol] : 0.0f;
#pragma unroll
  for (int r = 0; r < 8; ++r) {
    const int row = mt * 16 + r + ((lane >> 4) << 3);
    float v = c0[r] + c1[r] + bv;
    if (ACT) v = (v > THRESH) ? v : 0.0f;
    C[(size_t)row * N + bcol] = v;
  }
}

// ---------------------------------------------------------------------------
// Kernel: init h buffers + barrier counter
// ---------------------------------------------------------------------------
__global__ void init_kernel(float* __restrict__ hbufs, unsigned* __restrict__ counter) {
  const int i = blockIdx.x * blockDim.x + threadIdx.x;
  if (i < 2 * DB * DH) hbufs[i] = 0.0f;
  if (i == 0) *counter = 0u;
}

// ---------------------------------------------------------------------------
// Persistent GRU scan kernel.
//   16 workgroups; WG `bid` owns H-columns [bid*32, bid*32+32).
//   W_hh slice (3 gates x 32 cols x 512 = 192KB) resident in LDS (320KB WGP LDS)
//   for the whole 128-step scan, staged via async global->LDS B128.
//   Per step: async-stage h (64KB) + x_proj slice (12KB) to LDS, 12 waves each
//   compute one 16x16 gh tile via fp32 WMMA, gate math from LDS, device barrier.
// ---------------------------------------------------------------------------
#define NWG        16
#define GRUT       384          // 12 waves
#define HP         (DH + 4)     // padded pitch -> conflict-free ds_load_b64
#define KP         (DH + 4)
#define LDSW_FLTS  (96 * KP)                  // W_hh slice
#define LDSH_FLTS  (DB * HP)                  // h
#define LDSG_FLTS  (3 * DB * 32)              // gh tiles
#define LDSX_FLTS  (3 * DB * 32)              // x_proj slice for this step
#define GRU_LDS_BYTES ((LDSW_FLTS + LDSH_FLTS + LDSG_FLTS + LDSX_FLTS) * sizeof(float))

__global__ void gru_scan_kernel(const float* __restrict__ W_hh,
                                const float* __restrict__ b_hh,
                                const float* __restrict__ x_proj,
                                float* __restrict__ hbuf0,
                                float* __restrict__ hbuf1,
                                unsigned* __restrict__ counter) {
  extern __shared__ float smem[];
  float* lds_w  = smem;                                      // [96][KP]
  float* lds_h  = smem + LDSW_FLTS;                          // [32][HP]
  float* lds_gh = smem + LDSW_FLTS + LDSH_FLTS;              // [3][32][32]
  float* lds_x  = smem + LDSW_FLTS + LDSH_FLTS + LDSG_FLTS;  // [3][32][32]

  const unsigned lw_base = (unsigned)(uintptr_t)lds_w;
  const unsigned lh_base = (unsigned)(uintptr_t)lds_h;
  const unsigned lx_base = (unsigned)(uintptr_t)lds_x;

  const int tid  = threadIdx.x;
  const int bid  = blockIdx.x;
  const int lane = tid & 31;
  const int wave = tid >> 5;                  // 0..11
  const int col0 = bid * 32;                  // first H-column owned by this WG

  // ---- stage W_hh slice into LDS once via async B128 (rows: gate g, col c) ----
  for (int idx = tid; idx < (96 * DH) / 4; idx += GRUT) {
    const int r  = idx >> 7;                  // / (512/4)
    const int kc = (idx & 127) << 2;
    const int g  = r >> 5;
    const int c  = r & 31;
    async_g2l_b128(lw_base + (unsigned)((r * KP + kc) * 4),
                   &W_hh[((size_t)(g * DH + col0 + c)) * DH + kc]);
  }
  wait_asynccnt0();
  __syncthreads();

  // ---- this wave's tile: gate g, M-tile mt (batch), N-tile nt (cols) ----
  const int g   = wave >> 2;                  // 0..2
  const int mt  = (wave >> 1) & 1;
  const int nt  = wave & 1;
  const int l16 = lane & 15;
  const int kh  = (lane >> 4) << 1;

  const float gbias = b_hh[g * DH + col0 + nt * 16 + l16];

  const float* hin  = hbuf0;
  float*       hout = hbuf1;

  for (int t = 0; t < DS_; ++t) {
    // async-stage this step's x_proj slice: [3 gates][32 batch][32 cols]
    for (int idx = tid; idx < 3 * DB * 8; idx += GRUT) {
      const int q  = idx & 7;                 // float4 within the 32-col row
      const int b_ = (idx >> 3) & 31;
      const int gg = idx >> 8;
      async_g2l_b128(lx_base + (unsigned)((((gg * DB + b_) * 32) + q * 4) * 4),
                     &x_proj[((size_t)t * DB + b_) * G3 + gg * DH + col0 + q * 4]);
    }
    // async-stage h (B x H) into LDS
    for (int idx = tid; idx < (DB * DH) / 4; idx += GRUT) {
      const int r  = idx >> 7;
      const int kc = (idx & 127) << 2;
      async_g2l_b128(lh_base + (unsigned)((r * HP + kc) * 4),
                     &hin[(size_t)r * DH + kc]);
    }
    wait_asynccnt0();
    __syncthreads();

    // gh tile = h(16x512) @ W_hh_slice^T(512x16), fp32 WMMA, 2 acc chains
    const float* arow = &lds_h[(mt * 16 + l16) * HP + kh];
    const float* brow = &lds_w[(g * 32 + nt * 16 + l16) * KP + kh];
    v8f c0 = {}; v8f c1 = {};
#pragma unroll 4
    for (int k = 0; k < DH; k += 8) {
      v2f a0 = { arow[k],     arow[k + 1] };
      v2f b0 = { brow[k],     brow[k + 1] };
      c0 = wmma_f32(a0, b0, c0);
      v2f a1 = { arow[k + 4], arow[k + 5] };
      v2f b1 = { brow[k + 4], brow[k + 5] };
      c1 = wmma_f32(a1, b1, c1);
    }

    // write gh tile (+b_hh) to LDS: lds_gh[g][batch][col]
#pragma unroll
    for (int r = 0; r < 8; ++r) {
      const int row = mt * 16 + r + ((lane >> 4) << 3);
      const int cc  = nt * 16 + l16;
      lds_gh[(g * DB + row) * 32 + cc] = c0[r] + c1[r] + gbias;
    }
    __syncthreads();

    // gate math + h update for this WG's 32 columns (all operands in LDS)
    for (int idx = tid; idx < DB * 32; idx += GRUT) {
      const int b_i = idx >> 5;
      const int c_i = idx & 31;
      const int col = col0 + c_i;
      const float ghr = lds_gh[(0 * DB + b_i) * 32 + c_i];
      const float ghz = lds_gh[(1 * DB + b_i) * 32 + c_i];
      const float ghn = lds_gh[(2 * DB + b_i) * 32 + c_i];
      const float xr  = lds_x[(0 * DB + b_i) * 32 + c_i];
      const float xz  = lds_x[(1 * DB + b_i) * 32 + c_i];
      const float xn  = lds_x[(2 * DB + b_i) * 32 + c_i];
      const float r = 1.0f / (1.0f + expf(-(xr + ghr)));
      const float z = 1.0f / (1.0f + expf(-(xz + ghz)));
      const float n = tanhf(xn + r * ghn);
      const float hold = lds_h[b_i * HP + col];
      hout[(size_t)b_i * DH + col] = (1.0f - z) * n + z * hold;
    }
    __syncthreads();

    // device-wide barrier across the 16 co-resident WGs
    if (tid == 0) {
      __threadfence();
      atomicAdd(counter, 1u);
      const unsigned target = (unsigned)(NWG * (t + 1));
      while (__hip_atomic_load(counter, __ATOMIC_ACQUIRE,
                               __HIP_MEMORY_SCOPE_AGENT) < target) {
        __builtin_amdgcn_s_sleep(1);
      }
    }
    __syncthreads();

    // swap double buffer
    const float* tmp = hin; hin = hout; hout = (float*)tmp;
  }
}

// ---------------------------------------------------------------------------
// Host launcher
// ---------------------------------------------------------------------------
extern "C" void kernel_launch(void* const* d_in, const int* in_sizes, int n_in,
                              void* d_out, int out_size, void* d_ws, size_t ws_size,
                              hipStream_t stream) {
  const int*   tok  = (const int*)  d_in[0];
  const float* W1   = (const float*)d_in[1];
  const float* b1   = (const float*)d_in[2];
  const float* W2   = (const float*)d_in[3];
  const float* b2   = (const float*)d_in[4];
  const float* W_ih = (const float*)d_in[5];
  const float* W_hh = (const float*)d_in[6];
  const float* b_ih = (const float*)d_in[7];
  const float* b_hh = (const float*)d_in[8];
  const float* W3   = (const float*)d_in[9];
  const float* b3   = (const float*)d_in[10];
  float* out = (float*)d_out;

  // workspace layout (floats)
  float* ws = (float*)d_ws;
  float*   e1    = ws;                                   // 4096*512
  float*   emb   = e1    + (size_t)SB * DEH;             // 4096*256
  float*   xproj = emb   + (size_t)SB * DEO;             // 4096*1536
  float*   hbuf  = xproj + (size_t)SB * G3;              // 2 * 32*512
  unsigned* counter = (unsigned*)(hbuf + 2 * DB * DH);

  // 0) zero h0 + barrier counter
  init_kernel<<<(2 * DB * DH + 255) / 256, 256, 0, stream>>>(hbuf, counter);

  // 1) embedding gather + b1 + threshold
  gather_embed_kernel<<<SB, 128, 0, stream>>>(tok, W1, b1, e1);

  // 2) emb = thr(e1 @ W2 + b2)            (4096x512)@(512x256)
  {
    const int tiles = (SB / 16) * (DEO / 16);
    gemm_wmma_kernel<0, 1><<<(tiles + 7) / 8, 256, 0, stream>>>(
        e1, W2, b2, emb, SB, DEO, DEH);
  }

  // 3) x_proj = emb @ W_ih^T + b_ih       (4096x256)@(256x1536)
  {
    const int tiles = (SB / 16) * (G3 / 16);
    gemm_wmma_kernel<1, 0><<<(tiles + 7) / 8, 256, 0, stream>>>(
        emb, W_ih, b_ih, xproj, SB, G3, DEO);
  }

  // 4) GRU scan (persistent, 16 WGs, W_hh resident in 320KB LDS)
  (void)hipFuncSetAttribute((const void*)gru_scan_kernel,
                            hipFuncAttributeMaxDynamicSharedMemorySize,
                            (int)GRU_LDS_BYTES);
  gru_scan_kernel<<<NWG, GRUT, GRU_LDS_BYTES, stream>>>(
      W_hh, b_hh, xproj, hbuf, hbuf + DB * DH, counter);

  // 5) head: out = h_T @ W3 + b3          (32x512)@(512x256)
  // after 128 steps the final h is back in buffer 0
  {
    const int tiles = (DB / 16) * (DO_ / 16);
    gemm_wmma_kernel<0, 0><<<(tiles + 7) / 8, 256, 0, stream>>>(
        hbuf, W3, b3, out, DB, DO_, DH);
  }
}